// CustomGNNLayer4_91319594647967
// MI455X (gfx1250) — compile-verified
//
#include <hip/hip_runtime.h>
#include <math.h>

typedef float v2f __attribute__((ext_vector_type(2)));
typedef float v8f __attribute__((ext_vector_type(8)));
typedef unsigned int u32x4 __attribute__((ext_vector_type(4)));
typedef int i32x4 __attribute__((ext_vector_type(4)));
typedef int i32x8 __attribute__((ext_vector_type(8)));

#define N_NODES 64
#define FIN 256
#define FOUT 256
#define M_BIG 4096          // N*N
#define EPS_T4 1e-6f
#define BN_EPS 1e-5f

// workspace offsets (in floats)
enum {
  WS_R     = 0,        // 64   : r_i = sum_k H[i,k]^2
  WS_HSUM  = 64,       // 64   : hsum_i
  WS_COLSQ = 128,      // 256  : colsq_k = sum_i H[i,k]^2
  WS_SC    = 384,      // 16   : sc[0]=S_B=sum colsq^2, sc[1]=||hsum||^2
  WS_G     = 512,      // 4096 : H @ H^T
  WS_WN    = 4608,     // 4096 : Wn (== w, row-major vec)
  WS_U     = 8704,     // 4096 : Q upper coefs
  WS_DLT   = 12800,    // 4096 : Q diagonal
  WS_LOW   = 16896,    // 4096 : Q lower coefs
  WS_Y     = 20992,    // 64*4096 : sum_part_n @ (Q/64)
  WS_B     = 283136,   // 4096*256: normalized B
  WS_Z     = 1331712,  // 64*256 : Y @ B
  WS_PH    = 1348096,  // 64*256 : H@Wmlp^T + b
  WS_PX    = 1364480,  // 64*256 : X@Wmlp^T + b
  WS_ACC   = 1380864   // 64*256 : term accumulation
};

// ---------------------------------------------------------------------------
// Stats: row sums / row sq-sums of H, column sq-sums, Frobenius helpers.
// ---------------------------------------------------------------------------
__global__ void k_stats(const float* __restrict__ H, float* __restrict__ ws) {
  __shared__ float red[256];
  const int t = threadIdx.x;
  float cs = 0.f;
  for (int i = 0; i < N_NODES; ++i) { float h = H[i * FIN + t]; cs += h * h; }
  ws[WS_COLSQ + t] = cs;
  if (t < N_NODES) {
    float rr = 0.f, hs = 0.f;
    for (int k = 0; k < FIN; ++k) { float h = H[t * FIN + k]; rr += h * h; hs += h; }
    ws[WS_R + t] = rr;
    ws[WS_HSUM + t] = hs;
  }
  __syncthreads();
  red[t] = cs * cs;                       // S_B = ||B||_F^2
  __syncthreads();
  for (int s = 128; s > 0; s >>= 1) { if (t < s) red[t] += red[t + s]; __syncthreads(); }
  if (t == 0) ws[WS_SC + 0] = red[0];
  __syncthreads();
  float hv = (t < N_NODES) ? ws[WS_HSUM + t] : 0.f;
  red[t] = hv * hv;                       // ||hsum||^2
  __syncthreads();
  for (int s = 128; s > 0; s >>= 1) { if (t < s) red[t] += red[t + s]; __syncthreads(); }
  if (t == 0) ws[WS_SC + 1] = red[0];
}

// ---------------------------------------------------------------------------
// Branch-free fp32 WMMA GEMM (compile-time variants): one wave -> 16x16 tile of
//   C = (ACCUM ? C : 0) + alpha * A(MxK) * op(B) + (HAS_BIAS ? bias : 0)
// op(B): TRANSB ? B is (N,K) row-major (weight layout) : B is (K,N) row-major.
// Lane layouts per CDNA5 ISA 7.12.2 (32-bit A 16x4, 32-bit C/D 16x16).
// ---------------------------------------------------------------------------
template <int TRANSB, bool HAS_BIAS, bool ACCUM>
__global__ void k_gemm(const float* __restrict__ A, int lda,
                       const float* __restrict__ B, int ldb,
                       float* __restrict__ C, int ldc, int K,
                       const float* __restrict__ bias, float alpha) {
  const int lane = threadIdx.x & 31;
  const int half = lane >> 4;       // 0: lanes 0-15, 1: lanes 16-31
  const int l    = lane & 15;
  const int m0 = blockIdx.x * 16;
  const int n0 = blockIdx.y * 16;
  v8f c = {0.f, 0.f, 0.f, 0.f, 0.f, 0.f, 0.f, 0.f};
  const float* __restrict__ arow = A + (long)(m0 + l) * lda;
  const float* __restrict__ brow = TRANSB ? (B + (long)(n0 + l) * ldb)
                                          : (B + n0 + l);
#pragma unroll 4
  for (int k0 = 0; k0 < K; k0 += 4) {
    const int ka = k0 + 2 * half;   // VGPR v of A/B holds K = k0 + v + 2*half
    v2f a; a.x = arow[ka]; a.y = arow[ka + 1];
    v2f b;
    if (TRANSB) { b.x = brow[ka]; b.y = brow[ka + 1]; }
    else        { b.x = brow[(long)ka * ldb];
                  b.y = brow[(long)(ka + 1) * ldb]; }
    // speculative prefetch (OOB prefetches are dropped silently; no guard)
    __builtin_prefetch(arow + k0 + 64, 0, 1);            // global_prefetch_b8
    if (TRANSB) __builtin_prefetch(brow + k0 + 64, 0, 1);
    // D = A*B + C   (v_wmma_f32_16x16x4_f32)
    c = __builtin_amdgcn_wmma_f32_16x16x4_f32(false, a, false, b, (short)0, c,
                                              false, false);
  }
  const float bs = HAS_BIAS ? bias[n0 + l] : 0.f;
#pragma unroll
  for (int r = 0; r < 8; ++r) {
    const int row = m0 + r + 8 * half;   // C/D: VGPR r, lanes 16-31 -> M = r+8
    const long idx = (long)row * ldc + n0 + l;
    float v = alpha * c[r] + bs;
    if (ACCUM) v += C[idx];
    C[idx] = v;
  }
}

// ---------------------------------------------------------------------------
// Affinity + row softmax: Wn[j,t] over the 64x64 Gram matrix.
// ---------------------------------------------------------------------------
__global__ void k_softmax(const float* __restrict__ G, const float* __restrict__ rsq,
                          float* __restrict__ Wn) {
  __shared__ float red[64];
  const int j = blockIdx.x, t = threadIdx.x;
  const float inv2sig = 1.0f / (2.0f * 0.25f + 1e-7f);   // SCALE=0.5 -> sig2=0.25
  float D = rsq[j] - 2.0f * G[j * 64 + t] + rsq[t];
  D = fmaxf(D, 0.f);
  float Ds = fminf(D * inv2sig, 100.f);
  float val = fmaxf(expf(-Ds), 1e-7f) + 1e-7f;
  red[t] = val; __syncthreads();
  for (int s = 32; s > 0; s >>= 1) { if (t < s) red[t] = fmaxf(red[t], red[t + s]); __syncthreads(); }
  float mx = red[0]; __syncthreads();
  float e = expf(val - mx);
  red[t] = e; __syncthreads();
  for (int s = 32; s > 0; s >>= 1) { if (t < s) red[t] += red[t + s]; __syncthreads(); }
  Wn[j * 64 + t] = e / red[0];
}

// ---------------------------------------------------------------------------
// Closed-form Q of QR(T4): prefix-scan w^2 (Hillis-Steele over 1024 partials),
// then per-column semiseparable coefficients (u_k, delta_k, l_k), normalized.
// ---------------------------------------------------------------------------
__global__ void k_scanq(const float* __restrict__ w, float* __restrict__ ws) {
  __shared__ float bufA[1024];
  __shared__ float bufB[1024];
  const int t = threadIdx.x;
  float lw[4], ls[4], lsum = 0.f;
  for (int i = 0; i < 4; ++i) { lw[i] = w[t * 4 + i]; ls[i] = lw[i] * lw[i]; lsum += ls[i]; }
  bufA[t] = lsum; __syncthreads();
  float* src = bufA; float* dst = bufB;
  for (int off = 1; off < 1024; off <<= 1) {
    float v = src[t]; if (t >= off) v += src[t - off];
    dst[t] = v; __syncthreads();
    float* tmp = src; src = dst; dst = tmp;
  }
  const float S_w = src[1023];
  float P = (t == 0) ? 0.f : src[t - 1];   // exclusive prefix of w^2
  const float eps = EPS_T4;
  const float c = S_w + 2.f * eps;
  for (int i = 0; i < 4; ++i) {
    const int k = t * 4 + i;
    const float wk = lw[i];
    const float gamma = wk * (eps * eps) / (eps * eps + c * P);
    const float u = gamma * (1.f - c / eps);
    const float d = eps + gamma * wk;
    const float lo = gamma;
    float rest = S_w - P - wk * wk; if (rest < 0.f) rest = 0.f;
    const float inv = 1.f / sqrtf(u * u * P + d * d + lo * lo * rest);
    ws[WS_U + k] = u * inv;
    ws[WS_DLT + k] = d * inv;
    ws[WS_LOW + k] = lo * inv;
    P += ls[i];
  }
}

// ---------------------------------------------------------------------------
// Y = fro(sum_part) @ (Q / ||Q||_F): row j of sum_part is hsum/(8||hsum||)
// in block j; apply semiseparable Q with an O(1) piecewise prefix.
// ---------------------------------------------------------------------------
__global__ void k_applyQ(const float* __restrict__ w, const float* __restrict__ ws,
                         float* __restrict__ Y) {
  __shared__ float PS[65];
  __shared__ float ht[64];
  const int j = blockIdx.x, t = threadIdx.x;
  if (t == 0) {
    const float hn = 16.f * sqrtf(ws[WS_SC + 1]);        // ||sum_part||_F
    const float inv = (hn == 0.f) ? 1.f : 1.f / hn;
    float acc = 0.f;
    for (int r = 0; r < 64; ++r) {
      const float h = 2.f * ws[WS_HSUM + r] * inv;
      ht[r] = h;
      PS[r] = acc;
      acc += h * w[j * 64 + r];
    }
    PS[64] = acc;                                        // T_j
  }
  __syncthreads();
  const float T = PS[64];
  const int i0 = j * 64, i1 = i0 + 64;
  for (int k = t; k < M_BIG; k += 256) {
    const float pre = (k < i0) ? 0.f : ((k >= i1) ? T : PS[k - i0]);
    const float xk = (k >= i0 && k < i1) ? ht[k - i0] : 0.f;
    const float y = ws[WS_U + k] * pre + ws[WS_DLT + k] * xk +
                    ws[WS_LOW + k] * (T - pre - xk * w[k]);
    Y[(long)j * M_BIG + k] = y * (1.0f / 64.0f);         // ||Q||_F = 64 exactly
  }
}

// ---------------------------------------------------------------------------
// TDM helper: DMA one 256-float row from global memory into LDS.
// Builds a D# per CDNA5 ISA ch.8: group0 = {count=1, lds_addr, global_addr,
// type=2}; group1 = {mask=0, data_size=4B, tensor_dim0=tile_dim0=256,
// tensor_dim1=tile_dim1=1, dim0_stride=256}. Groups 2/3 unused (2D tensor).
// ---------------------------------------------------------------------------
#if __has_builtin(__builtin_amdgcn_tensor_load_to_lds)
#define HAVE_TDM 1
__device__ __forceinline__ void tdm_load_row(unsigned int lds_addr,
                                             unsigned long long gaddr) {
  u32x4 g0;
  g0.x = 1u;                                       // count=1, is_restore=0
  g0.y = lds_addr;                                 // lds_addr (bytes)
  g0.z = (unsigned int)gaddr;                      // global_addr[31:0]
  g0.w = (unsigned int)((gaddr >> 32) & 0x1FFFFFFull) | (2u << 30); // [56:32]|type=2
  i32x8 g1;
  g1[0] = 0x00020000;                              // data_size=2 (4 bytes), mask=0
  g1[1] = (int)(256u << 16);                       // tensor_dim0[15:0]=256 @bit48
  g1[2] = (int)(1u << 16);                         // tensor_dim0 hi=0, tensor_dim1=1
  g1[3] = (int)(256u << 16);                       // tensor_dim1 hi=0, tile_dim0=256
  g1[4] = 1;                                       // tile_dim1=1, tile_dim2=0
  g1[5] = 256;                                     // tensor_dim0_stride=256
  g1[6] = 0;                                       // stride hi / dim1_stride lo
  g1[7] = 0;
  i32x4 z4 = {0, 0, 0, 0};
#if defined(__clang_major__) && (__clang_major__ >= 23)
  i32x8 z8 = {0, 0, 0, 0, 0, 0, 0, 0};
  __builtin_amdgcn_tensor_load_to_lds(g0, g1, z4, z4, z8, 0);
#else
  __builtin_amdgcn_tensor_load_to_lds(g0, g1, z4, z4, 0);
#endif
}
#else
#define HAVE_TDM 0
#endif

// ---------------------------------------------------------------------------
// B[(i,r),k] = H[i,k]*H[r,k] / ||B||_F, with H rows staged to LDS by the TDM.
// ---------------------------------------------------------------------------
__global__ void k_buildB(const float* __restrict__ H, const float* __restrict__ ws,
                         float* __restrict__ Bm) {
  __shared__ float sh[2 * FIN];
  const int p = blockIdx.x;
  const int i = p >> 6, r = p & 63;
  const int t = threadIdx.x;
#if HAVE_TDM
  if (t < 32) {                                     // one wave issues the DMA
    // LDS byte offset = low 32 bits of the generic pointer (ISA 10.2)
    const unsigned int lds0 = (unsigned int)(unsigned long long)(void*)sh;
    tdm_load_row(lds0, (unsigned long long)(const void*)(H + (long)i * FIN));
    tdm_load_row(lds0 + FIN * 4,
                 (unsigned long long)(const void*)(H + (long)r * FIN));
    __builtin_amdgcn_s_wait_tensorcnt(0);
  }
  __syncthreads();
#else
  sh[t] = H[(long)i * FIN + t];
  sh[FIN + t] = H[(long)r * FIN + t];
  __syncthreads();
#endif
  const float nb = sqrtf(ws[WS_SC + 0]);
  const float inv = (nb == 0.f) ? 1.f : 1.f / nb;
  Bm[(long)p * FIN + t] = sh[t] * sh[FIN + t] * inv;
}

// ---------------------------------------------------------------------------
// out = BN(acc + P_H): biased variance over the 64 rows, per column.
// ---------------------------------------------------------------------------
__global__ void k_bn(const float* __restrict__ acc, const float* __restrict__ PH,
                     const float* __restrict__ gamma, const float* __restrict__ beta,
                     float* __restrict__ out) {
  const int f = threadIdx.x;
  float s = 0.f, s2 = 0.f;
  for (int n = 0; n < N_NODES; ++n) {
    const float v = acc[n * FOUT + f] + PH[n * FOUT + f];
    s += v; s2 += v * v;
  }
  const float mu = s * (1.f / 64.f);
  float var = s2 * (1.f / 64.f) - mu * mu;
  if (var < 0.f) var = 0.f;
  const float g = gamma[f] / sqrtf(var + BN_EPS);
  const float b = beta[f];
  for (int n = 0; n < N_NODES; ++n) {
    const float v = acc[n * FOUT + f] + PH[n * FOUT + f];
    out[n * FOUT + f] = (v - mu) * g + b;
  }
}

// ---------------------------------------------------------------------------
extern "C" void kernel_launch(void* const* d_in, const int* in_sizes, int n_in,
                              void* d_out, int out_size, void* d_ws, size_t ws_size,
                              hipStream_t stream) {
  (void)in_sizes; (void)n_in; (void)out_size; (void)ws_size;
  const float* H  = (const float*)d_in[0];
  const float* X  = (const float*)d_in[1];
  const float* Wm = (const float*)d_in[2];   // (Fout, Fin)
  const float* bm = (const float*)d_in[3];
  const float* Wa = (const float*)d_in[4];   // (Fout, Fout)
  const float* ba = (const float*)d_in[5];
  const float* Wb = (const float*)d_in[6];   // (Fout, Fin=Fout)
  const float* bb = (const float*)d_in[7];
  const float* gg = (const float*)d_in[8];
  const float* gb = (const float*)d_in[9];
  float* ws  = (float*)d_ws;
  float* out = (float*)d_out;

  k_stats<<<1, 256, 0, stream>>>(H, ws);
  // G = H @ H^T  (64x64, K=256)
  k_gemm<1, false, false><<<dim3(4, 4), 32, 0, stream>>>(
      H, FIN, H, FIN, ws + WS_G, 64, FIN, nullptr, 1.f);
  k_softmax<<<64, 64, 0, stream>>>(ws + WS_G, ws + WS_R, ws + WS_WN);
  k_scanq<<<1, 1024, 0, stream>>>(ws + WS_WN, ws);
  k_applyQ<<<64, 256, 0, stream>>>(ws + WS_WN, ws, ws + WS_Y);
  k_buildB<<<4096, 256, 0, stream>>>(H, ws, ws + WS_B);
  // Z = Y @ B  (64x256, K=4096)
  k_gemm<0, false, false><<<dim3(4, 16), 32, 0, stream>>>(
      ws + WS_Y, M_BIG, ws + WS_B, FOUT, ws + WS_Z, FOUT, M_BIG, nullptr, 1.f);
  // P_H = H @ Wmlp^T + b ; P_X = X @ Wmlp^T + b
  k_gemm<1, true, false><<<dim3(4, 16), 32, 0, stream>>>(
      H, FIN, Wm, FIN, ws + WS_PH, FOUT, FIN, bm, 1.f);
  k_gemm<1, true, false><<<dim3(4, 16), 32, 0, stream>>>(
      X, FIN, Wm, FIN, ws + WS_PX, FOUT, FIN, bm, 1.f);
  // acc  = P_X @ Wa^T + ba            (term2)
  k_gemm<1, true, false><<<dim3(4, 16), 32, 0, stream>>>(
      ws + WS_PX, FOUT, Wa, FOUT, ws + WS_ACC, FOUT, FOUT, ba, 1.f);
  // acc -= P_H @ Wa                   (term1 minus its P_H part; +P_H in BN)
  k_gemm<0, false, true><<<dim3(4, 16), 32, 0, stream>>>(
      ws + WS_PH, FOUT, Wa, FOUT, ws + WS_ACC, FOUT, FOUT, nullptr, -1.f);
  // acc += Z @ Wb^T + bb              (term3)
  k_gemm<1, true, true><<<dim3(4, 16), 32, 0, stream>>>(
      ws + WS_Z, FOUT, Wb, FOUT, ws + WS_ACC, FOUT, FOUT, bb, 1.f);
  k_bn<<<1, 256, 0, stream>>>(ws + WS_ACC, ws + WS_PH, gg, gb, out);
}